// NaiveBiasBase_36876589204101
// MI455X (gfx1250) — compile-verified
//
#include <hip/hip_runtime.h>
#include <cstdint>
#include <cstddef>

// ---------------------------------------------------------------------------
// bias[h, j, i] = exp(w[h, (S-L) + j - i] - offset[h])
//   w: [H, S] with S = 2L-1,  out: [H, L, L] float32
//
// Store-bound kernel (256 MB out vs 262 KB in). Strategy:
//   - one 128x128 tile per workgroup; its Toeplitz window is 255 floats
//   - async-copy raw window global->LDS (gfx1250 ASYNCcnt path), exp in place
//   - diagonal shift in registers: 1 new ds_load per float4 stored
//   - coalesced non-temporal global_store_b128
// ---------------------------------------------------------------------------

#define TJ 128
#define TI 128
#define WSZ (TJ + TI - 1)          // 255 valid window entries (pad LDS to 256)

typedef __attribute__((address_space(1))) int* gas_i32;   // global int*
typedef __attribute__((address_space(3))) int* lds_i32;   // LDS int*
typedef float vfloat4 __attribute__((ext_vector_type(4)));

__global__ __launch_bounds__(256) void toeplitz_bias_exp_kernel(
    const float* __restrict__ w,
    const float* __restrict__ offset,
    float* __restrict__ out,
    int L, int table)
{
    __shared__ float pwin[WSZ + 1];          // +1 pad: harmless overfetch on last shift

    const int i0 = blockIdx.x * TI;
    const int j0 = blockIdx.y * TJ;
    const int h  = blockIdx.z;

    // window start in p-index space: min idx over tile = (S-L) + j0 - (i0+TI-1)
    const int g0 = (table - L) + j0 - (i0 + TI - 1);

    const int t = threadIdx.x;

    // ---- Phase 1a: async copy raw w window into LDS (ASYNCcnt path) -------
    const float* gsrc = w + (size_t)h * (size_t)table + (size_t)g0;
#if __has_builtin(__builtin_amdgcn_global_load_async_to_lds_b32)
    if (t < WSZ) {
        __builtin_amdgcn_global_load_async_to_lds_b32(
            (gas_i32)(gsrc + t), (lds_i32)&pwin[t], /*offset=*/0, /*cpol=*/0);
    }
#else
    if (t < WSZ) {
        uint32_t ldsaddr = (uint32_t)(uintptr_t)(lds_i32)&pwin[t];
        uint32_t voff    = (uint32_t)(t * 4);
        uint64_t sbase   = (uint64_t)(uintptr_t)gsrc;
        asm volatile("global_load_async_to_lds_b32 %0, %1, %2"
                     :: "v"(ldsaddr), "v"(voff), "s"(sbase) : "memory");
    }
#endif
#if __has_builtin(__builtin_amdgcn_s_wait_asynccnt)
    __builtin_amdgcn_s_wait_asynccnt(0);
#else
    asm volatile("s_wait_asynccnt 0" ::: "memory");
#endif
    __syncthreads();

    // ---- Phase 1b: exp in place (each thread touches only its own slot) ---
    const float off = offset[h];
    if (t < WSZ) pwin[t] = expf(pwin[t] - off);
    __syncthreads();

    // ---- Phase 2: write the 128x128 tile ----------------------------------
    // Thread map: lane (t&31) -> 4-column group, (t>>5) -> block of 16 rows.
    // out(jj, ii..ii+3) = { pwin[127+jj-ii], pwin[126+jj-ii],
    //                       pwin[125+jj-ii], pwin[124+jj-ii] }
    // With base = (TI-4) + jj0 - ii and r = jj - jj0:
    //   v = (pwin[base+3+r], pwin[base+2+r], pwin[base+1+r], pwin[base+r])
    // so each row needs just one new LDS read after the first four.
    const int ii       = (t & 31) * 4;
    const int jj0      = (t >> 5) * (TJ / 8);        // 16 rows per thread
    const int rows_per = TJ / 8;

    const int base = (TI - 4) + jj0 - ii;            // in [0, 236]
    float x0 = pwin[base + 0];
    float x1 = pwin[base + 1];
    float x2 = pwin[base + 2];
    float x3 = pwin[base + 3];

    const size_t Ls = (size_t)L;
    vfloat4* outp = (vfloat4*)(out + ((size_t)h * Ls + (size_t)(j0 + jj0)) * Ls
                                    + (size_t)(i0 + ii));
    const size_t rowstride4 = Ls / 4;

    #pragma unroll
    for (int r = 0; r < rows_per; ++r) {
        vfloat4 v = {x3, x2, x1, x0};
        __builtin_nontemporal_store(v, outp);
        outp += rowstride4;
        x0 = x1; x1 = x2; x2 = x3;
        x3 = pwin[base + 4 + r];                      // max index 255 (padded)
    }
}

extern "C" void kernel_launch(void* const* d_in, const int* in_sizes, int n_in,
                              void* d_out, int out_size, void* d_ws, size_t ws_size,
                              hipStream_t stream) {
    (void)n_in; (void)out_size; (void)d_ws; (void)ws_size;

    const float* w      = (const float*)d_in[0];   // [H, 2L-1] f32
    const float* offset = (const float*)d_in[1];   // [H] f32
    float*       out    = (float*)d_out;           // [H, L, L] f32

    const int H     = in_sizes[1];
    const int table = in_sizes[0] / H;             // 2L-1
    const int L     = (table + 1) / 2;             // 2048

    dim3 grid(L / TI, L / TJ, H);                  // 16 x 16 x 16
    toeplitz_bias_exp_kernel<<<grid, 256, 0, stream>>>(w, offset, out, L, table);
}